// MaskedDiffusion_82076825027303
// MI455X (gfx1250) — compile-verified
//
#include <hip/hip_runtime.h>
#include <hip/hip_bf16.h>
#include <stdint.h>

#define BB 256
#define NN 65536
#define PI_F 3.14159265358979323846f

// ---------------------------------------------------------------------------
// CDNA5 async copy helpers (gfx1250): global -> LDS via the async engine.
// Tracked with ASYNCcnt; LDS dst address is the low 32 bits of the generic
// pointer (LDS aperture maps low bits to the wave's LDS offset).
// ---------------------------------------------------------------------------
__device__ __forceinline__ void async_copy16(void* lds_dst, const void* gsrc) {
    uint32_t l = (uint32_t)(uintptr_t)lds_dst;
    asm volatile("global_load_async_to_lds_b128 %0, %1, off"
                 :: "v"(l), "v"(gsrc) : "memory");
}
__device__ __forceinline__ void wait_async_le2() {
    asm volatile("s_wait_asynccnt 2" ::: "memory");
}
__device__ __forceinline__ void wait_async_0() {
    asm volatile("s_wait_asynccnt 0" ::: "memory");
}
__device__ __forceinline__ void wait_ds_0() {
    asm volatile("s_wait_dscnt 0" ::: "memory");
}

// float -> order-preserving uint32 (larger float => larger key)
__device__ __forceinline__ uint32_t fkey(float x) {
    uint32_t u = __float_as_uint(x);
    return (u & 0x80000000u) ? ~u : (u | 0x80000000u);
}

__device__ __forceinline__ uint32_t wkey(float u, float d) {
    float g = -logf(-logf(u + 1e-7f) + 1e-7f);   // gumbel(U)
    return fkey(g + logf(d));                    // + log dirichlet
}

// ---------------------------------------------------------------------------
// K0: zero the histogram region of the workspace (5 MB) each launch
// ---------------------------------------------------------------------------
__global__ void k_zero(uint4* p, unsigned n4) {
    unsigned i = blockIdx.x * blockDim.x + threadIdx.x;
    if (i < n4) { uint4 z; z.x = z.y = z.z = z.w = 0u; p[i] = z; }
}

// ---------------------------------------------------------------------------
// K1: per-row k and loss weight w
// ---------------------------------------------------------------------------
__global__ void k_init(const float* __restrict__ t_in, uint32_t* __restrict__ ks,
                       float* __restrict__ w_out) {
    int b = blockIdx.x * blockDim.x + threadIdx.x;
    if (b < BB) {
        float tv = t_in[b];
        float r = 1.0f - cosf(0.5f * PI_F * tv);            // cosine_schedule
        int k = (int)((float)NN * r);                       // trunc like astype(int32)
        k = k < 0 ? 0 : (k > NN ? NN : k);
        ks[b] = (uint32_t)k;
        float ta = tv * (1.0f - 2e-3f) + 1e-3f;             // cosine_weight
        w_out[b] = 0.5f * PI_F * sinf(0.5f * PI_F * ta);
    }
}

// ---------------------------------------------------------------------------
// K2: map U,D -> keys, plus level-1 histogram of key[31:21] (2048 bins).
// Block = 256 threads, covers 8192 contiguous elements of one row.
// U/D tiles staged via the async copy engine, double buffered: the async
// engine keeps the next tile in flight (ASYNCcnt=2) while the logf chains run.
// ---------------------------------------------------------------------------
__global__ void k_map_hist1(const float* __restrict__ U, const float* __restrict__ D,
                            uint32_t* __restrict__ keys, uint32_t* __restrict__ hist1) {
    const int t = threadIdx.x;
    const int row = blockIdx.x >> 3;
    const int chunk = blockIdx.x & 7;
    const size_t rowBase = (size_t)row * NN;
    const int base = chunk * 8192;

    __shared__ float sU[2][1024];
    __shared__ float sD[2][1024];
    __shared__ uint32_t shist[2048];
    for (int i = t; i < 2048; i += 256) shist[i] = 0u;
    __syncthreads();

    const float* gU = U + rowBase + base;
    const float* gD = D + rowBase + base;

    // prologue: tile 0 into buffer 0
    async_copy16(&sU[0][t * 4], gU + t * 4);
    async_copy16(&sD[0][t * 4], gD + t * 4);

    for (int it = 0; it < 8; ++it) {
        const int cb = it & 1;
        if (it < 7) {
            const int nb = cb ^ 1;
            wait_ds_0();   // prior ds_reads of buffer nb fully drained
            async_copy16(&sU[nb][t * 4], gU + (it + 1) * 1024 + t * 4);
            async_copy16(&sD[nb][t * 4], gD + (it + 1) * 1024 + t * 4);
            wait_async_le2();   // current tile (2 older ops) landed; next in flight
        } else {
            wait_async_0();
        }
        float4 u4 = *(const float4*)&sU[cb][t * 4];
        float4 d4 = *(const float4*)&sD[cb][t * 4];
        uint4 kv;
        kv.x = wkey(u4.x, d4.x);
        kv.y = wkey(u4.y, d4.y);
        kv.z = wkey(u4.z, d4.z);
        kv.w = wkey(u4.w, d4.w);
        *(uint4*)(keys + rowBase + base + it * 1024 + t * 4) = kv;
        atomicAdd(&shist[kv.x >> 21], 1u);
        atomicAdd(&shist[kv.y >> 21], 1u);
        atomicAdd(&shist[kv.z >> 21], 1u);
        atomicAdd(&shist[kv.w >> 21], 1u);
    }
    __syncthreads();
    for (int i = t; i < 2048; i += 256) {
        uint32_t v = shist[i];
        if (v) atomicAdd(&hist1[row * 2048 + i], v);
    }
}

// ---------------------------------------------------------------------------
// K3: per-row select over a histogram (descending suffix scan).
// mode 1: in=ks            -> prefOut = T1 (11b),           kOut = k1
// mode 2: in=k1, pref=T1   -> prefOut = (T1<<11)|T2 (22b),  kOut = k2
// mode 3: in=k2, pref=22b  -> thrKey = (pref<<10)|T3, tieTake = k3,
//                             eqCnt = exact # of keys equal to thrKey
// Sentinel prefOut = 0xFFFFFFFF when k == 0 (select nothing).
// ---------------------------------------------------------------------------
__global__ void k_select(const uint32_t* __restrict__ hist, int bins,
                         const uint32_t* __restrict__ kin,
                         const uint32_t* __restrict__ prefIn,
                         uint32_t* __restrict__ prefOut, uint32_t* __restrict__ kOut,
                         uint32_t* __restrict__ thrKey, uint32_t* __restrict__ tieTake,
                         uint32_t* __restrict__ eqCnt, int mode) {
    const int b = blockIdx.x;
    const int t = threadIdx.x;
    __shared__ uint32_t part[256];
    __shared__ uint32_t sT, sK;

    uint32_t k = kin[b];
    bool skip = (mode == 1) ? (k == 0u)
                            : (prefIn[b] == 0xFFFFFFFFu || k == 0u);
    if (skip) {
        if (t == 0) {
            if (mode == 3) { thrKey[b] = 0xFFFFFFFFu; tieTake[b] = 0u; eqCnt[b] = 0u; }
            else           { prefOut[b] = 0xFFFFFFFFu; kOut[b] = 0u; }
        }
        return;
    }
    if (t == 0) { sT = 0u; sK = 0u; }

    const uint32_t* h = hist + (size_t)b * bins;
    const int chunk = bins / 256;
    uint32_t s = 0;
    for (int j = 0; j < chunk; ++j) s += h[bins - 1 - (t * chunk + j)];
    part[t] = s;
    __syncthreads();
    if (t == 0) {   // serial exclusive scan over 256 partials (trivial work)
        uint32_t run = 0;
        for (int i = 0; i < 256; ++i) { uint32_t v = part[i]; part[i] = run; run += v; }
    }
    __syncthreads();
    uint32_t run = part[t];
    for (int j = 0; j < chunk; ++j) {
        int bin = bins - 1 - (t * chunk + j);
        uint32_t v = h[bin];
        if (run < k && run + v >= k) { sT = (uint32_t)bin; sK = k - run; }
        run += v;
    }
    __syncthreads();
    if (t == 0) {
        if (mode == 1)      { prefOut[b] = sT;                     kOut[b] = sK; }
        else if (mode == 2) { prefOut[b] = (prefIn[b] << 11) | sT; kOut[b] = sK; }
        else {
            thrKey[b]  = (prefIn[b] << 10) | sT;
            tieTake[b] = sK;
            eqCnt[b]   = h[sT];   // exact count of keys == thrKey in this row
        }
    }
}

// ---------------------------------------------------------------------------
// K4: refinement histograms, LDS-privatized (hot level-1 bins can hold >10% of
// a row; ds_add_u32 absorbs the conflicts, global atomics only on flush of
// nonzero bins). level 2: key[20:10] within T1 bin; level 3: key[9:0] within
// the 22-bit prefix. Block = 256 threads over 8192 contiguous row elements.
// ---------------------------------------------------------------------------
__global__ void k_refine(const uint32_t* __restrict__ keys,
                         const uint32_t* __restrict__ prefIn,
                         uint32_t* __restrict__ histOut, int level) {
    const int t = threadIdx.x;
    const int row = blockIdx.x >> 3;
    const uint32_t pref = prefIn[row];
    if (pref == 0xFFFFFFFFu) return;          // k==0 row: nothing to refine
    const int chunk = blockIdx.x & 7;
    const size_t base = (size_t)row * NN + chunk * 8192;
    const int bins = (level == 2) ? 2048 : 1024;

    __shared__ uint32_t shist[2048];
    for (int i = t; i < bins; i += 256) shist[i] = 0u;
    __syncthreads();

    for (int it = 0; it < 8; ++it) {
        uint4 kv = *(const uint4*)(keys + base + it * 1024 + t * 4);
        if (level == 2) {
            if ((kv.x >> 21) == pref) atomicAdd(&shist[(kv.x >> 10) & 0x7FFu], 1u);
            if ((kv.y >> 21) == pref) atomicAdd(&shist[(kv.y >> 10) & 0x7FFu], 1u);
            if ((kv.z >> 21) == pref) atomicAdd(&shist[(kv.z >> 10) & 0x7FFu], 1u);
            if ((kv.w >> 21) == pref) atomicAdd(&shist[(kv.w >> 10) & 0x7FFu], 1u);
        } else {
            if ((kv.x >> 10) == pref) atomicAdd(&shist[kv.x & 0x3FFu], 1u);
            if ((kv.y >> 10) == pref) atomicAdd(&shist[kv.y & 0x3FFu], 1u);
            if ((kv.z >> 10) == pref) atomicAdd(&shist[kv.z & 0x3FFu], 1u);
            if ((kv.w >> 10) == pref) atomicAdd(&shist[kv.w & 0x3FFu], 1u);
        }
    }
    __syncthreads();
    uint32_t* h = histOut + (size_t)row * bins;
    for (int i = t; i < bins; i += 256) {
        uint32_t v = shist[i];
        if (v) atomicAdd(&h[i], v);
    }
}

// ---------------------------------------------------------------------------
// K5: deterministic per-block tie counts — ONLY for rows with excess ties
// (eqCnt > tieTake). Common-case rows early-exit without touching keys.
// ---------------------------------------------------------------------------
__global__ void k_tie_count(const uint32_t* __restrict__ keys,
                            const uint32_t* __restrict__ thrKey,
                            const uint32_t* __restrict__ tieTake,
                            const uint32_t* __restrict__ eqCnt,
                            uint32_t* __restrict__ blkCnt) {
    const int t = threadIdx.x;
    const int row = blockIdx.x >> 3;
    if (eqCnt[row] == tieTake[row]) return;   // uniform branch: no rank pass needed
    const int chunk = blockIdx.x & 7;
    const uint32_t thr = thrKey[row];
    const size_t base = (size_t)row * NN + chunk * 8192;
    uint32_t c = 0;
    for (int it = 0; it < 8; ++it) {
        uint4 kv = *(const uint4*)(keys + base + it * 1024 + t * 4);
        c += (kv.x == thr) + (kv.y == thr) + (kv.z == thr) + (kv.w == thr);
    }
    __shared__ uint32_t red[256];
    red[t] = c;
    __syncthreads();
    for (int s = 128; s > 0; s >>= 1) { if (t < s) red[t] += red[t + s]; __syncthreads(); }
    if (t == 0) blkCnt[blockIdx.x] = red[0];
}

// ---------------------------------------------------------------------------
// K6: final mask.
// Fast path (eqCnt == tieTake): mask = key > thr, plus ALL ties (take != 0).
// Rare path (excess ties): first tieTake ties in INDEX order via deterministic
// block prefix counts + in-block scans (matches jnp stable argsort tie-break).
// ---------------------------------------------------------------------------
__global__ void k_final_mask(const uint32_t* __restrict__ keys,
                             const uint32_t* __restrict__ thrKey,
                             const uint32_t* __restrict__ tieTake,
                             const uint32_t* __restrict__ eqCnt,
                             const uint32_t* __restrict__ blkCnt,
                             float* __restrict__ out) {
    const int t = threadIdx.x;
    const int row = blockIdx.x >> 3;
    const int chunk = blockIdx.x & 7;
    const uint32_t thr = thrKey[row];
    const uint32_t take = tieTake[row];
    const size_t base = (size_t)row * NN + chunk * 8192;

    if (eqCnt[row] == take) {
        // streaming fast path: no LDS, no scans
        const bool anyTie = (take != 0u);
        for (int it = 0; it < 8; ++it) {
            uint4 kv = *(const uint4*)(keys + base + it * 1024 + t * 4);
            float4 m;
            m.x = ((kv.x > thr) || (anyTie && kv.x == thr)) ? 1.0f : 0.0f;
            m.y = ((kv.y > thr) || (anyTie && kv.y == thr)) ? 1.0f : 0.0f;
            m.z = ((kv.z > thr) || (anyTie && kv.z == thr)) ? 1.0f : 0.0f;
            m.w = ((kv.w > thr) || (anyTie && kv.w == thr)) ? 1.0f : 0.0f;
            *(float4*)(out + base + it * 1024 + t * 4) = m;
        }
        return;
    }

    // rare path: deterministic index-ordered tie ranks
    __shared__ uint32_t scan[256];
    __shared__ uint32_t sBase;
    if (t == 0) {
        uint32_t s = 0;
        for (int c = 0; c < chunk; ++c) s += blkCnt[(row << 3) + c];
        sBase = s;
    }
    __syncthreads();
    for (int it = 0; it < 8; ++it) {
        uint4 kv = *(const uint4*)(keys + base + it * 1024 + t * 4);
        uint32_t e0 = kv.x == thr, e1 = kv.y == thr, e2 = kv.z == thr, e3 = kv.w == thr;
        scan[t] = e0 + e1 + e2 + e3;
        __syncthreads();
        if (t == 0) {
            uint32_t run = sBase;
            for (int i = 0; i < 256; ++i) { uint32_t v = scan[i]; scan[i] = run; run += v; }
            sBase = run;
        }
        __syncthreads();
        uint32_t r = scan[t];
        float4 m;
        m.x = ((kv.x > thr) || (e0 && r < take)) ? 1.0f : 0.0f; r += e0;
        m.y = ((kv.y > thr) || (e1 && r < take)) ? 1.0f : 0.0f; r += e1;
        m.z = ((kv.z > thr) || (e2 && r < take)) ? 1.0f : 0.0f; r += e2;
        m.w = ((kv.w > thr) || (e3 && r < take)) ? 1.0f : 0.0f;
        *(float4*)(out + base + it * 1024 + t * 4) = m;
        __syncthreads();  // scan[]/sBase reused next iteration
    }
}

// ---------------------------------------------------------------------------
// Host side
// ---------------------------------------------------------------------------
extern "C" void kernel_launch(void* const* d_in, const int* in_sizes, int n_in,
                              void* d_out, int out_size, void* d_ws, size_t ws_size,
                              hipStream_t stream) {
    (void)in_sizes; (void)n_in; (void)out_size; (void)ws_size;
    const float* t_in = (const float*)d_in[1];   // d_in[0] = batch (unused)
    const float* U    = (const float*)d_in[2];
    const float* D    = (const float*)d_in[3];
    float* out        = (float*)d_out;           // [B*N mask floats][B weights]

    // workspace layout
    uint8_t*  ws    = (uint8_t*)d_ws;
    uint32_t* keys  = (uint32_t*)ws;                               // 64 MB
    uint32_t* hist1 = (uint32_t*)(ws + (size_t)BB * NN * 4);       // 2 MB
    uint32_t* hist2 = hist1 + (size_t)BB * 2048;                   // 2 MB
    uint32_t* hist3 = hist2 + (size_t)BB * 2048;                   // 1 MB
    uint32_t* st    = hist3 + (size_t)BB * 1024;
    uint32_t* ks     = st;
    uint32_t* pref1  = st + BB;
    uint32_t* k1     = st + 2 * BB;
    uint32_t* pref2  = st + 3 * BB;
    uint32_t* k2     = st + 4 * BB;
    uint32_t* thrKey = st + 5 * BB;
    uint32_t* tieTk  = st + 6 * BB;
    uint32_t* eqCnt  = st + 7 * BB;
    uint32_t* blkCnt = st + 8 * BB;                                // B*8

    // zero hist1..hist3 (5 MB = 327680 uint4)
    const unsigned n4 = (BB * 2048 + BB * 2048 + BB * 1024) / 4;
    k_zero<<<(n4 + 255) / 256, 256, 0, stream>>>((uint4*)hist1, n4);
    k_init<<<1, BB, 0, stream>>>(t_in, ks, out + (size_t)BB * NN);
    k_map_hist1<<<BB * 8, 256, 0, stream>>>(U, D, keys, hist1);
    k_select<<<BB, 256, 0, stream>>>(hist1, 2048, ks, nullptr, pref1, k1,
                                     nullptr, nullptr, nullptr, 1);
    k_refine<<<BB * 8, 256, 0, stream>>>(keys, pref1, hist2, 2);
    k_select<<<BB, 256, 0, stream>>>(hist2, 2048, k1, pref1, pref2, k2,
                                     nullptr, nullptr, nullptr, 2);
    k_refine<<<BB * 8, 256, 0, stream>>>(keys, pref2, hist3, 3);
    k_select<<<BB, 256, 0, stream>>>(hist3, 1024, k2, pref2, nullptr, nullptr,
                                     thrKey, tieTk, eqCnt, 3);
    k_tie_count<<<BB * 8, 256, 0, stream>>>(keys, thrKey, tieTk, eqCnt, blkCnt);
    k_final_mask<<<BB * 8, 256, 0, stream>>>(keys, thrKey, tieTk, eqCnt, blkCnt, out);
}